// VGAE_62697932587536
// MI455X (gfx1250) — compile-verified
//
#include <hip/hip_runtime.h>

#define N_NODES 10000
#define N_EDGES 320000
#define F_IN    128
#define HID     32
#define NTILES  625   // 10000 / 16

typedef __attribute__((ext_vector_type(2))) float v2f;
typedef __attribute__((ext_vector_type(8))) float v8f;

// ---------------- zero scratch ----------------
__global__ void vgae_zero(float* __restrict__ p, int n) {
    int i = blockIdx.x * blockDim.x + threadIdx.x;
    if (i < n) p[i] = 0.0f;
}

// ------- layer-1 scatter: agg1[dst] += x[src], deg[dst] += 1 -------
// one wave (32 lanes) per edge; lane handles features f = lane, lane+32, ...
__global__ void __launch_bounds__(256)
vgae_scatter1(const float* __restrict__ x, const int* __restrict__ src,
              const int* __restrict__ dst, float* __restrict__ agg1,
              float* __restrict__ deg) {
    int t = blockIdx.x * blockDim.x + threadIdx.x;
    int e = t >> 5;
    int lane = t & 31;
    if (e >= N_EDGES) return;
    int s = src[e], d = dst[e];
    #pragma unroll
    for (int f = lane; f < F_IN; f += 32)
        atomicAdd(&agg1[(size_t)d * F_IN + f], x[(size_t)s * F_IN + f]);
    if (lane == 0) atomicAdd(&deg[d], 1.0f);
}

// ------- layer 1: h = relu(x @ W1s + (agg1/deg) @ W1n + b1) -------
// thread = (node, j); node uniform per 32-thread group, j = lane
__global__ void __launch_bounds__(256)
vgae_layer1(const float* __restrict__ x, const float* __restrict__ agg1,
            const float* __restrict__ deg, const float* __restrict__ Ws,
            const float* __restrict__ Wn, const float* __restrict__ b,
            float* __restrict__ h) {
    int t = blockIdx.x * blockDim.x + threadIdx.x;
    if (t >= N_NODES * HID) return;
    int node = t >> 5;
    int j = t & 31;
    float inv = 1.0f / fmaxf(deg[node], 1.0f);
    float accx = 0.0f, accn = 0.0f;
    const float* xr = x    + (size_t)node * F_IN;
    const float* ar = agg1 + (size_t)node * F_IN;
    for (int k = 0; k < F_IN; ++k) {
        accx = fmaf(xr[k], Ws[k * HID + j], accx);
        accn = fmaf(ar[k], Wn[k * HID + j], accn);
    }
    float v = accx + inv * accn + b[j];
    h[t] = fmaxf(v, 0.0f);
}

// ------- layer-2 scatter: agg2[dst] += h[src]  (HID==32: lane per feature) ----
__global__ void __launch_bounds__(256)
vgae_scatter2(const float* __restrict__ h, const int* __restrict__ src,
              const int* __restrict__ dst, float* __restrict__ agg2) {
    int t = blockIdx.x * blockDim.x + threadIdx.x;
    int e = t >> 5;
    int f = t & 31;
    if (e >= N_EDGES) return;
    atomicAdd(&agg2[(size_t)dst[e] * HID + f], h[(size_t)src[e] * HID + f]);
}

// ------- layer 2: mu, logstd, z = mu + eps*exp(logstd) -------
__global__ void __launch_bounds__(256)
vgae_layer2(const float* __restrict__ h, const float* __restrict__ agg2,
            const float* __restrict__ deg,
            const float* __restrict__ Wms, const float* __restrict__ Wmn,
            const float* __restrict__ bmu,
            const float* __restrict__ Wls, const float* __restrict__ Wln,
            const float* __restrict__ bls,
            const float* __restrict__ eps,
            float* __restrict__ mu_out, float* __restrict__ ls_out,
            float* __restrict__ z) {
    int t = blockIdx.x * blockDim.x + threadIdx.x;
    if (t >= N_NODES * HID) return;
    int node = t >> 5;
    int j = t & 31;
    float inv = 1.0f / fmaxf(deg[node], 1.0f);
    const float* hr = h    + (size_t)node * HID;
    const float* ar = agg2 + (size_t)node * HID;
    float ms = 0.0f, mn = 0.0f, ls = 0.0f, ln = 0.0f;
    #pragma unroll 8
    for (int k = 0; k < HID; ++k) {
        float hv = hr[k];
        float av = ar[k];
        ms = fmaf(hv, Wms[k * HID + j], ms);
        mn = fmaf(av, Wmn[k * HID + j], mn);
        ls = fmaf(hv, Wls[k * HID + j], ls);
        ln = fmaf(av, Wln[k * HID + j], ln);
    }
    float mu = ms + inv * mn + bmu[j];
    float lg = ls + inv * ln + bls[j];
    mu_out[t] = mu;
    ls_out[t] = lg;
    z[t] = fmaf(eps[t], __expf(lg), mu);
}

// fast sigmoid: 1 TRANS + 2 VALU per element (avoid IEEE div expansion)
__device__ __forceinline__ float fast_sigmoid(float x) {
#if __has_builtin(__builtin_amdgcn_tanhf)
    // sigmoid(x) = 0.5 * tanh(x/2) + 0.5  -> v_tanh_f32 (single TRANS32)
    return fmaf(0.5f, __builtin_amdgcn_tanhf(x * 0.5f), 0.5f);
#else
    // v_exp_f32 + v_rcp_f32, no v_div_scale/div_fixup sequence
    return __builtin_amdgcn_rcpf(1.0f + __expf(-x));
#endif
}

// ------- decode: adj = sigmoid(z @ z^T) via V_WMMA_F32_16X16X4_F32 -------
// One wave computes one 16x16 tile. adj[m][n] = sum_k z[m][k]*z[n][k].
// f32 A (16x4) lane layout: row M = lane&15, K = 2*(lane>>4) + {0,1} -> float2.
// f32 B (4x16) lane layout: col N = lane&15, K = 2*(lane>>4) + {0,1} -> same
// load pattern as A (B[k][n] = z[n][k]), so A and B loads are identical shape.
__global__ void __launch_bounds__(256)
vgae_decode(const float* __restrict__ z, float* __restrict__ adj) {
    const int lane  = threadIdx.x & 31;
    const int wave  = threadIdx.x >> 5;
    const int tileN = blockIdx.x * 8 + wave;   // wave-uniform
    const int tileM = blockIdx.y;
    if (tileN >= NTILES) return;               // uniform per wave: EXEC stays full

    const int row0  = tileM * 16;
    const int col0  = tileN * 16;
    const int r     = lane & 15;
    const int khalf = lane >> 4;               // 0 or 1

    const float* ap = z + (size_t)(row0 + r) * HID + khalf * 2;
    const float* bp = z + (size_t)(col0 + r) * HID + khalf * 2;

    v8f c = {};
    #pragma unroll
    for (int kk = 0; kk < 8; ++kk) {           // K = 32 in steps of 4
        v2f a = *(const v2f*)(ap + kk * 4);
        v2f b = *(const v2f*)(bp + kk * 4);
        c = __builtin_amdgcn_wmma_f32_16x16x4_f32(
                /*neg_a=*/false, a, /*neg_b=*/false, b,
                /*c_mod=*/(short)0, c, /*reuse_a=*/false, /*reuse_b=*/false);
    }

    // C/D layout: VGPR v -> M = v + 8*khalf, N = lane&15
    const int mbase = row0 + 8 * khalf;
    #pragma unroll
    for (int v = 0; v < 8; ++v) {
        adj[(size_t)(mbase + v) * N_NODES + col0 + r] = fast_sigmoid(c[v]);
    }
}

extern "C" void kernel_launch(void* const* d_in, const int* in_sizes, int n_in,
                              void* d_out, int out_size, void* d_ws, size_t ws_size,
                              hipStream_t stream) {
    const float* x   = (const float*)d_in[0];
    const int*   src = (const int*)  d_in[1];
    const int*   dst = (const int*)  d_in[2];
    const float* eps = (const float*)d_in[3];
    const float* W1s = (const float*)d_in[4];
    const float* W1n = (const float*)d_in[5];
    const float* b1  = (const float*)d_in[6];
    const float* Wms = (const float*)d_in[7];
    const float* Wmn = (const float*)d_in[8];
    const float* bmu = (const float*)d_in[9];
    const float* Wls = (const float*)d_in[10];
    const float* Wln = (const float*)d_in[11];
    const float* bls = (const float*)d_in[12];

    float* out    = (float*)d_out;
    float* adj    = out;                                   // [N, N]
    float* mu_out = out + (size_t)N_NODES * N_NODES;       // [N, HID]
    float* ls_out = mu_out + (size_t)N_NODES * HID;        // [N, HID]

    float* ws   = (float*)d_ws;
    float* agg1 = ws;                                      // N*F_IN
    float* deg  = agg1 + (size_t)N_NODES * F_IN;           // N
    float* h    = deg  + N_NODES;                          // N*HID
    float* agg2 = h    + (size_t)N_NODES * HID;            // N*HID
    float* z    = agg2 + (size_t)N_NODES * HID;            // N*HID

    // zero agg1 + deg + h + agg2 span (h overwritten anyway; deterministic)
    const int nz = N_NODES * F_IN + N_NODES + 2 * N_NODES * HID;
    vgae_zero<<<(nz + 255) / 256, 256, 0, stream>>>(agg1, nz);

    // layer-1 neighbor aggregation
    const int et = N_EDGES * 32;
    vgae_scatter1<<<(et + 255) / 256, 256, 0, stream>>>(x, src, dst, agg1, deg);

    // layer 1 dense + relu
    const int nt = N_NODES * HID;
    vgae_layer1<<<(nt + 255) / 256, 256, 0, stream>>>(x, agg1, deg, W1s, W1n, b1, h);

    // layer-2 neighbor aggregation
    vgae_scatter2<<<(et + 255) / 256, 256, 0, stream>>>(h, src, dst, agg2);

    // layer 2 dense (mu, logstd) + reparameterize
    vgae_layer2<<<(nt + 255) / 256, 256, 0, stream>>>(h, agg2, deg,
                                                     Wms, Wmn, bmu,
                                                     Wls, Wln, bls,
                                                     eps, mu_out, ls_out, z);

    // decode: sigmoid(z z^T) with fp32 WMMA, 8 waves/block, tile = 16x16
    dim3 grid((NTILES + 7) / 8, NTILES);
    vgae_decode<<<grid, 256, 0, stream>>>(z, adj);
}